// NeuronClassPyTorch_48215302865320
// MI455X (gfx1250) — compile-verified
//
#include <hip/hip_runtime.h>
#include <hip/hip_bf16.h>
#include <cstddef>
#include <cstdint>

// ---------------- problem constants (from reference) ----------------
constexpr int   T_  = 128;
constexpr int   B_  = 256;
constexpr int   N_  = 4096;
constexpr int   TB_ = T_ * B_;                 // 32768 rows of the matvec
constexpr float ALPHA   = 1.0f - 0.05f / 10.0f; // 0.995
constexpr float GAMMA_  = 0.95f;
constexpr float MU_     = 0.1f;
constexpr float SIGMA_P = 0.15f;
constexpr float V_TH    = 1.0f;
// R_BASE = 1 + exp(-0.5*(sqrt(2)/1)^2) = 1 + e^-1
constexpr float R_BASE  = 1.36787944117144233f;

typedef __attribute__((ext_vector_type(2))) float        v2f;
typedef __attribute__((ext_vector_type(8))) float        v8f;
typedef __attribute__((ext_vector_type(4))) unsigned int u32x4;
typedef __attribute__((ext_vector_type(8))) int          i32x8;
typedef __attribute__((ext_vector_type(4))) int          i32x4;

#if defined(__has_builtin)
#if __has_builtin(__builtin_amdgcn_tensor_load_to_lds)
#define HAVE_TDM 1
#else
#define HAVE_TDM 0
#endif
#else
#define HAVE_TDM 0
#endif

// =====================================================================
// Kernel 1: S[row] = dot(x[row, :], w)  for row in [0, T*B), x as [TB, N]
// WMMA f32 16x16x4, 16 rows per wave. A tiles (16 x 64 f32, row stride N)
// are staged global->LDS by the Tensor Data Mover with hardware row
// padding (64 dwords + 1 pad dword == bank-conflict-free [64+1] layout),
// double-buffered per wave and synchronized with s_wait_tensorcnt.
// B-matrix is the w chunk in column 0; column 0 of D = the 16 dots.
// =====================================================================
#define S_WAVES 4
#define TILE_K  64

__global__ __launch_bounds__(S_WAVES * 32)
void k_matvec_wmma(const float* __restrict__ x, const float* __restrict__ w,
                   float* __restrict__ S) {
  __shared__ float tile[S_WAVES][2][16][TILE_K + 1]; // [64+1]: conflict-free columns
  const int lane    = threadIdx.x & 31;
  const int wv      = __builtin_amdgcn_readfirstlane(threadIdx.x >> 5);
  const int rowbase = (blockIdx.x * S_WAVES + wv) * 16;
  const int r       = lane & 15;            // matrix row within fragment
  const int kof     = (lane >> 4) << 1;     // lanes 0-15 -> K 0,1 ; lanes 16-31 -> K 2,3

  const unsigned long long gbase =
      (unsigned long long)(uintptr_t)x + (unsigned long long)rowbase * N_ * 4ull;

  // ---- issue one 16 x TILE_K tile load into LDS buffer `buf` at column k0
  auto issue = [&](int buf, int k0) {
#if HAVE_TDM
    const unsigned int lds_off =
        (unsigned int)(uintptr_t)(&tile[wv][buf][0][0]);     // low 32 bits = LDS offset
    const unsigned long long ga = gbase + (unsigned long long)k0 * 4ull;
    u32x4 g0;
    g0[0] = 1u;                                              // count=1, load, no gather
    g0[1] = lds_off;                                         // lds_addr
    g0[2] = (unsigned int)(ga & 0xffffffffu);                // global_addr[31:0]
    g0[3] = (unsigned int)((ga >> 32) & 0x01ffffffu)         // global_addr[56:32]
            | (2u << 30);                                    // type = 2 (image)
    i32x8 g1 = {};
    g1[0] = (2 << 16)        // data_size = 4 bytes
          | (1 << 20)        // pad_enable
          | (5 << 22);       // pad_interval: 2^(5+1) = 64 dwords
                             // pad_amount = 0 -> 1 dword pad per row
    g1[1] = (int)((unsigned)N_ << 16);          // tensor_dim0[15:0] = 4096
    g1[2] = (int)(16u << 16);                   // tensor_dim0[31:16]=0, tensor_dim1=16
    g1[3] = (int)((unsigned)TILE_K << 16);      // tile_dim0 = 64
    g1[4] = 16;                                 // tile_dim1 = 16, tile_dim2 = 0
    g1[5] = N_;                                 // tensor_dim0_stride = 4096 elements
    g1[6] = 0;
    g1[7] = 0;
    const i32x4 z4 = {};
#if __clang_major__ >= 23
    const i32x8 z8 = {};
    __builtin_amdgcn_tensor_load_to_lds(g0, g1, z4, z4, z8, 0);
#else
    __builtin_amdgcn_tensor_load_to_lds(g0, g1, z4, z4, 0);
#endif
#else
    // fallback: per-lane coalesced float4 staging
#pragma unroll
    for (int i = 0; i < (16 * TILE_K / 4) / 32; ++i) {
      const int slot = i * 32 + lane;
      const int rr   = slot >> 4;
      const int cc   = (slot & 15) << 2;
      const float4 v = *(const float4*)&x[(size_t)(rowbase + rr) * N_ + k0 + cc];
      tile[wv][buf][rr][cc + 0] = v.x;
      tile[wv][buf][rr][cc + 1] = v.y;
      tile[wv][buf][rr][cc + 2] = v.z;
      tile[wv][buf][rr][cc + 3] = v.w;
    }
#endif
  };

  v8f acc = {};
  issue(0, 0);
  for (int k0 = 0; k0 < N_; k0 += TILE_K) {
    const int buf = (k0 / TILE_K) & 1;
    if (k0 + TILE_K < N_) {
      issue(buf ^ 1, k0 + TILE_K);             // prefetch next tile via TDM
#if HAVE_TDM
      __builtin_amdgcn_s_wait_tensorcnt(1);    // current tile resident
#endif
    } else {
#if HAVE_TDM
      __builtin_amdgcn_s_wait_tensorcnt(0);
#endif
    }
    // ---- 16 WMMAs over this tile
#pragma unroll
    for (int kk = 0; kk < TILE_K; kk += 4) {
      v2f A;
      A.x = tile[wv][buf][r][kk + kof + 0];
      A.y = tile[wv][buf][r][kk + kof + 1];
      const float w0 = w[k0 + kk + kof + 0];
      const float w1 = w[k0 + kk + kof + 1];
      v2f Bv;
      Bv.x = (r == 0) ? w0 : 0.0f;   // one-hot column 0 holds the w chunk
      Bv.y = (r == 0) ? w1 : 0.0f;
      acc = __builtin_amdgcn_wmma_f32_16x16x4_f32(
          /*neg_a=*/false, A, /*neg_b=*/false, Bv,
          /*c_mod=*/(short)0, acc, /*reuse_a=*/false, /*reuse_b=*/false);
    }
  }
  // column 0 of D: lane 0 -> rows M=0..7 (VGPR j), lane 16 -> rows M=8..15
  if (r == 0) {
    const int mo = (lane >> 4) * 8;
#pragma unroll
    for (int j = 0; j < 8; ++j) S[rowbase + mo + j] = acc[j];
  }
}

// =====================================================================
// Kernel 2: per-(b,n) linear traces over t (streams x once, coalesced).
//   p     = alpha*p + x_t           (keep p at t=T-2 as pprev, final as pfin)
//   xiloc = gamma*xiloc + (1-gamma)*p    (xi[n] = mean_b xiloc later)
// =====================================================================
__global__ __launch_bounds__(256)
void k_traces(const float* __restrict__ x, float* __restrict__ pprev,
              float* __restrict__ pfin, float* __restrict__ xibuf) {
  const size_t gid  = (size_t)blockIdx.x * blockDim.x + threadIdx.x; // 0..B*N/4-1
  const size_t off4 = gid * 4;                                       // flat b*N+n
  const size_t slab = (size_t)B_ * N_;
  float4 p  = {0.f, 0.f, 0.f, 0.f};
  float4 xi = {0.f, 0.f, 0.f, 0.f};
  float4 pp = {0.f, 0.f, 0.f, 0.f};
  for (int t = 0; t < T_; ++t) {
    const float4 xv = *(const float4*)&x[(size_t)t * slab + off4];
    if (t + 1 < T_)
      __builtin_prefetch(&x[(size_t)(t + 1) * slab + off4], 0, 0); // global_prefetch_b8
    if (t == T_ - 1) pp = p;                 // p BEFORE the final update
    p.x = ALPHA * p.x + xv.x;  p.y = ALPHA * p.y + xv.y;
    p.z = ALPHA * p.z + xv.z;  p.w = ALPHA * p.w + xv.w;
    xi.x = GAMMA_ * xi.x + (1.0f - GAMMA_) * p.x;
    xi.y = GAMMA_ * xi.y + (1.0f - GAMMA_) * p.y;
    xi.z = GAMMA_ * xi.z + (1.0f - GAMMA_) * p.z;
    xi.w = GAMMA_ * xi.w + (1.0f - GAMMA_) * p.w;
  }
  *(float4*)&pprev[off4] = pp;
  *(float4*)&pfin[off4]  = p;
  *(float4*)&xibuf[off4] = xi;
}

// =====================================================================
// Kernel 3: the (B,) LIF scan over t (needs S). One block of 256 threads,
// one thread per batch row. Also emits E[b], R[b], v_final.
// =====================================================================
__global__ __launch_bounds__(256)
void k_scan(const float* __restrict__ S, const float* __restrict__ w,
            float* __restrict__ zs_out, float* __restrict__ v_out,
            float* __restrict__ E_out, float* __restrict__ R_out,
            float* __restrict__ vfin_ws) {
  __shared__ float red[256];
  const int b = threadIdx.x;
  // ||w||^2 block reduction
  float wsq = 0.f;
  for (int n = b; n < N_; n += 256) { const float ww = w[n]; wsq += ww * ww; }
  red[b] = wsq;
  __syncthreads();
  for (int s = 128; s > 0; s >>= 1) {
    if (b < s) red[b] += red[b + s];
    __syncthreads();
  }
  wsq = red[0];

  float v = 0.f, z = 0.f;
  for (int t = 0; t < T_; ++t) {
    v = ALPHA * v + S[t * B_ + b] - V_TH * z;
    z = (v - V_TH > 0.f) ? 1.f : 0.f;
    zs_out[t * B_ + b] = z;
  }
  v_out[b]   = v;
  vfin_ws[b] = v;
  // E = eps@w = dot(x_last,w) - v*||w||^2  (v after final update)
  E_out[b] = S[(T_ - 1) * B_ + b] - v * wsq;
  R_out[b] = (z > 0.f) ? R_BASE * 1.5f : R_BASE;
}

// =====================================================================
// Kernel 4: xi[n] = mean_b xibuf[b,n]
// =====================================================================
__global__ __launch_bounds__(256)
void k_xi(const float* __restrict__ xibuf, float* __restrict__ xi) {
  const int n = blockIdx.x * blockDim.x + threadIdx.x;
  float s = 0.f;
  for (int b = 0; b < B_; ++b) s += xibuf[(size_t)b * N_ + n];
  xi[n] = s * (1.0f / B_);
}

// =====================================================================
// Kernel 5: final-step grad. One block per b.
//   phase 1: Gamma[b] = mean_n exp(-0.5*((pfin - xi)/sigma)^2)
//   phase 2: grad = (-(v*eps + E*pprev) + mu*(1-Gamma)*pfin) * R
// =====================================================================
__global__ __launch_bounds__(256)
void k_grad(const float* __restrict__ xlast, const float* __restrict__ w,
            const float* __restrict__ pprev, const float* __restrict__ pfin,
            const float* __restrict__ xi, const float* __restrict__ E,
            const float* __restrict__ R, const float* __restrict__ vfin,
            float* __restrict__ grad) {
  __shared__ float red[256];
  const int b   = blockIdx.x;
  const int tid = threadIdx.x;
  const size_t base = (size_t)b * N_;

  float ss = 0.f;
  for (int n = tid; n < N_; n += 256) {
    const float d = (pfin[base + n] - xi[n]) * (1.0f / SIGMA_P);
    ss += __expf(-0.5f * d * d);
  }
  red[tid] = ss;
  __syncthreads();
  for (int s = 128; s > 0; s >>= 1) {
    if (tid < s) red[tid] += red[tid + s];
    __syncthreads();
  }
  const float coef = MU_ * (1.0f - red[0] * (1.0f / N_));
  const float vb = vfin[b], Eb = E[b], Rb = R[b];
  for (int n = tid; n < N_; n += 256) {
    const float eps = xlast[base + n] - vb * w[n];
    const float g   = -(vb * eps + Eb * pprev[base + n]) + coef * pfin[base + n];
    grad[base + n]  = g * Rb;
  }
}

// =====================================================================
extern "C" void kernel_launch(void* const* d_in, const int* in_sizes, int n_in,
                              void* d_out, int out_size, void* d_ws, size_t ws_size,
                              hipStream_t stream) {
  const float* x = (const float*)d_in[0];   // (T,B,N) f32
  const float* w = (const float*)d_in[1];   // (N,)    f32

  float* out  = (float*)d_out;
  float* zs   = out;                              // (T,B)
  float* grad = out + TB_;                        // (B,N)
  float* vout = out + TB_ + (size_t)B_ * N_;      // (B,)

  float* ws = (float*)d_ws;
  size_t o = 0;
  float* S     = ws + o; o += TB_;
  float* pprev = ws + o; o += (size_t)B_ * N_;
  float* pfin  = ws + o; o += (size_t)B_ * N_;
  float* xibuf = ws + o; o += (size_t)B_ * N_;
  float* xi    = ws + o; o += N_;
  float* E     = ws + o; o += B_;
  float* Rr    = ws + o; o += B_;
  float* vfin  = ws + o; o += B_;

  // S = X @ w  (WMMA matvec over the full [T*B, N] view of x, TDM-staged)
  k_matvec_wmma<<<TB_ / (16 * S_WAVES), S_WAVES * 32, 0, stream>>>(x, w, S);
  // per-(b,n) linear traces (independent of S)
  k_traces<<<((size_t)B_ * N_ / 4) / 256, 256, 0, stream>>>(x, pprev, pfin, xibuf);
  // sequential LIF scan + E/R/v_final
  k_scan<<<1, 256, 0, stream>>>(S, w, zs, vout, E, Rr, vfin);
  // xi = mean over batch
  k_xi<<<N_ / 256, 256, 0, stream>>>(xibuf, xi);
  // final-step gradient
  k_grad<<<B_, 256, 0, stream>>>(x + (size_t)(T_ - 1) * B_ * N_, w,
                                 pprev, pfin, xi, E, Rr, vfin, grad);
}